// KalmanFilter_76450417869034
// MI455X (gfx1250) — compile-verified
//
#include <hip/hip_runtime.h>
#include <stdint.h>

typedef __attribute__((ext_vector_type(2))) float v2f;
typedef __attribute__((ext_vector_type(8))) float v8f;

#define NB 256
#define NT 200

// ---------------------------------------------------------------------------
// Async global->LDS staging (gfx1250 GLOBAL_LOAD_ASYNC_TO_LDS_*, ASYNCcnt).
// Per-lane: each lane moves 16B (b128) or 8B (b64). Generic pointers to
// __shared__ truncate to the LDS byte offset (flat LDS aperture rule).
// ---------------------------------------------------------------------------
__device__ __forceinline__ void async_copy(float* __restrict__ dst,
                                           const float* __restrict__ src,
                                           int nbytes)
{
    const int lane = threadIdx.x & 31;
    const uint32_t l = (uint32_t)(uintptr_t)dst;
    const uint64_t g = (uint64_t)(uintptr_t)src;
    int off = 0;
    for (; off + 512 <= nbytes; off += 512) {     // 32 lanes * 16B
        asm volatile("global_load_async_to_lds_b128 %0, %1, off"
                     :: "v"(l + (uint32_t)off + (uint32_t)lane * 16u),
                        "v"(g + (uint64_t)off + (uint64_t)lane * 16u)
                     : "memory");
    }
    if (off < nbytes) {                           // 256B tail: 32 lanes * 8B
        asm volatile("global_load_async_to_lds_b64 %0, %1, off"
                     :: "v"(l + (uint32_t)off + (uint32_t)lane * 8u),
                        "v"(g + (uint64_t)off + (uint64_t)lane * 8u)
                     : "memory");
    }
}

__device__ __forceinline__ void wait_async0()
{
    asm volatile("s_wait_asynccnt 0x0" ::: "memory");
}

// ---------------------------------------------------------------------------
// D(16x16) = A(16x16) * op(B)(16x16) [+ Cadd], accumulated via f32 WMMA
// 16x16x4. Row-major LDS tiles, row stride 16. kmax trims known-zero K-range
// from zero-embedded operands. btrans=true: op(B)=B^T.
// ---------------------------------------------------------------------------
__device__ __forceinline__ void wmma_mm16(float* __restrict__ D,
                                          const float* __restrict__ A,
                                          const float* __restrict__ Bm,
                                          const float* __restrict__ Cadd,
                                          int kmax, bool btrans)
{
    const int lane = threadIdx.x & 31;
    const int lo = lane & 15;
    const int hi = lane >> 4;
    v8f acc;
#pragma unroll
    for (int v = 0; v < 8; ++v)
        acc[v] = Cadd ? Cadd[(v + 8 * hi) * 16 + lo] : 0.0f;

    for (int k0 = 0; k0 < kmax; k0 += 4) {
        const int ka = k0 + 2 * hi;
        v2f a, b;
        a.x = A[lo * 16 + ka];
        a.y = A[lo * 16 + ka + 1];
        if (btrans) { b.x = Bm[lo * 16 + ka]; b.y = Bm[lo * 16 + ka + 1]; }
        else        { b.x = Bm[ka * 16 + lo]; b.y = Bm[(ka + 1) * 16 + lo]; }
        acc = __builtin_amdgcn_wmma_f32_16x16x4_f32(
            false, a, false, b, (short)0, acc, false, false);
    }
#pragma unroll
    for (int v = 0; v < 8; ++v)
        D[(v + 8 * hi) * 16 + lo] = acc[v];
}

__global__ __launch_bounds__(32)
void KalmanFilter_gfx1250_kernel(const float* __restrict__ obs,    // [B,T,8]
                                 const float* __restrict__ ctrl,   // [B,T-1,4]
                                 const float* __restrict__ Amat,   // [B,T-1,16,16]
                                 const float* __restrict__ Bmat,   // [B,T-1,16,4]
                                 const float* __restrict__ Cmat,   // [B,T,8,16]
                                 const float* __restrict__ Qmat,   // [B,T-1,16,16]
                                 const float* __restrict__ Rmat,   // [B,T,8,8]
                                 const float* __restrict__ boff,   // [B,T-1,16]
                                 const float* __restrict__ doff,   // [B,T,8]
                                 const float* __restrict__ mu0,    // [B,16]
                                 const float* __restrict__ P0,     // [B,16,16]
                                 float* __restrict__ out)
{
    // double-buffered async staging targets
    __shared__ float Cm2[2][256], Rl2[2][64], Am2[2][256], Qm2[2][256], Bt2[2][64];
    // work tiles
    __shared__ float P[256], Pn[256], CP[256], Sf[256], Sinv[256],
                     PCt[256], Km[256], FC[256], Tm[256];
    __shared__ float Sa[128];
    __shared__ float pmv[16], fmv[16], btv[16], inn[8], dtv[8], ytv[8], utv[4];

    const int lane = threadIdx.x;
    const int b = blockIdx.x;

    // ---- init carry + zero embedding pads (Cm rows 8..15 in both buffers,
    //      Sinv everywhere; only its 8x8 top-left is rewritten per step)
    {
        const float4* p0 = (const float4*)(P0 + (size_t)b * 256);
        float4* p = (float4*)P;
        for (int i = lane; i < 64; i += 32) p[i] = p0[i];
    }
    for (int i = lane; i < 256; i += 32) {
        Cm2[0][i] = 0.0f;
        Cm2[1][i] = 0.0f;
        Sinv[i]   = 0.0f;
    }
    if (lane < 16) pmv[lane] = mu0[b * 16 + lane];

    // ---- prologue: stage step-0 operands asynchronously
    async_copy(Cm2[0], Cmat + (size_t)b * NT * 128, 512);
    async_copy(Rl2[0], Rmat + (size_t)b * NT * 64, 256);
    async_copy(Am2[0], Amat + (size_t)b * (NT - 1) * 256, 1024);
    async_copy(Qm2[0], Qmat + (size_t)b * (NT - 1) * 256, 1024);
    async_copy(Bt2[0], Bmat + (size_t)b * (NT - 1) * 64, 256);
    __syncthreads();

    float* fm_out = out;
    float* fc_out = out + (size_t)NB * NT * 16;
    float* pm_out = fc_out + (size_t)NB * NT * 256;
    float* pc_out = pm_out + (size_t)NB * NT * 16;

    for (int t = 0; t < NT; ++t) {
        const int buf = t & 1, nxt = buf ^ 1;
        const size_t ot = (size_t)b * NT + t;
        const size_t pt = (size_t)b * (NT - 1) + t;

        // staged data for step t is resident after this
        wait_async0();
        __syncthreads();

        // ---- kick off step t+1 staging; overlaps with all of step t's math
        if (t + 1 < NT) {
            async_copy(Cm2[nxt], Cmat + (ot + 1) * 128, 512);
            async_copy(Rl2[nxt], Rmat + (ot + 1) * 64, 256);
        }
        if (t + 1 < NT - 1) {
            async_copy(Am2[nxt], Amat + (pt + 1) * 256, 1024);
            async_copy(Qm2[nxt], Qmat + (pt + 1) * 256, 1024);
            async_copy(Bt2[nxt], Bmat + (pt + 1) * 64, 256);
        }
        // warm L2 one step further ahead so the async copies hit cache
        if (t + 2 < NT)     __builtin_prefetch(Cmat + (ot + 2) * 128, 0, 2);
        if (t + 2 < NT - 1) {
            __builtin_prefetch(Amat + (pt + 2) * 256, 0, 2);
            __builtin_prefetch(Qmat + (pt + 2) * 256, 0, 2);
        }

        const float* Cm = Cm2[buf];
        const float* Rl = Rl2[buf];
        const float* Am = Am2[buf];
        const float* Qm = Qm2[buf];
        const float* Bt = Bt2[buf];

        // ---- emit predicted (pre-update) outputs: carry at time t
        if (lane < 16) pm_out[ot * 16 + lane] = pmv[lane];
        {
            const float4* s = (const float4*)P;
            float4* d = (float4*)(pc_out + ot * 256);
            for (int i = lane; i < 64; i += 32) d[i] = s[i];
        }
        if (lane < 8) { dtv[lane] = doff[ot * 8 + lane]; ytv[lane] = obs[ot * 8 + lane]; }
        __syncthreads();

        // ---- CP = C*P ; Sf = CP*C^T  (C zero-embedded 16x16)
        wmma_mm16(CP, Cm, P, nullptr, 16, false);
        __syncthreads();
        wmma_mm16(Sf, CP, Cm, nullptr, 16, true);
        __syncthreads();

        // ---- invert S = Sf[0:8,0:8] + R via Gauss-Jordan on augmented [S|I]
        for (int i = lane; i < 128; i += 32) {
            int r = i >> 4, c = i & 15;
            Sa[i] = (c < 8) ? (Sf[r * 16 + c] + Rl[r * 8 + c])
                            : ((c - 8 == r) ? 1.0f : 0.0f);
        }
        __syncthreads();
        for (int k = 0; k < 8; ++k) {
            if (lane == k) {
                float inv = 1.0f / Sa[k * 16 + k];
                for (int j = 0; j < 16; ++j) Sa[k * 16 + j] *= inv;
            }
            __syncthreads();
            if (lane < 8 && lane != k) {
                float f = Sa[lane * 16 + k];
                for (int j = 0; j < 16; ++j) Sa[lane * 16 + j] -= f * Sa[k * 16 + j];
            }
            __syncthreads();
        }
        for (int i = lane; i < 64; i += 32) {
            int r = i >> 3, c = i & 7;
            Sinv[r * 16 + c] = Sa[r * 16 + 8 + c];
        }
        __syncthreads();

        // ---- K = P*C^T*Sinv  (Sinv rows 8..15 zero -> kmax=8 on 2nd mm)
        wmma_mm16(PCt, P, Cm, nullptr, 16, true);
        __syncthreads();
        wmma_mm16(Km, PCt, Sinv, nullptr, 8, false);
        __syncthreads();

        // ---- innovation + filtered mean
        if (lane < 8) {
            float po = dtv[lane];
            for (int j = 0; j < 16; ++j) po += Cm[lane * 16 + j] * pmv[j];
            inn[lane] = ytv[lane] - po;
        }
        __syncthreads();
        if (lane < 16) {
            float v = pmv[lane];
            for (int j = 0; j < 8; ++j) v += Km[lane * 16 + j] * inn[j];
            fmv[lane] = v;
        }
        __syncthreads();

        // ---- FC = P - K*(C*P)   (K cols 8..15 zero -> kmax=8)
        wmma_mm16(Tm, Km, CP, nullptr, 8, false);
        __syncthreads();
        for (int i = lane; i < 256; i += 32) FC[i] = P[i] - Tm[i];
        __syncthreads();

        // ---- emit filtered outputs
        if (lane < 16) fm_out[ot * 16 + lane] = fmv[lane];
        {
            const float4* s = (const float4*)FC;
            float4* d = (float4*)(fc_out + ot * 256);
            for (int i = lane; i < 64; i += 32) d[i] = s[i];
        }

        // ---- predict (skipped at last step; reference discards padded step)
        if (t + 1 < NT) {
            if (lane < 16) btv[lane] = boff[pt * 16 + lane];
            if (lane < 4)  utv[lane] = ctrl[pt * 4 + lane];
            __syncthreads();

            // Pn = A*FC*A^T + Q
            wmma_mm16(Tm, Am, FC, nullptr, 16, false);
            __syncthreads();
            wmma_mm16(Pn, Tm, Am, Qm, 16, true);
            __syncthreads();

            if (lane < 16) {
                float v = btv[lane];
                for (int j = 0; j < 16; ++j) v += Am[lane * 16 + j] * fmv[j];
                for (int j = 0; j < 4;  ++j) v += Bt[lane * 4 + j] * utv[j];
                pmv[lane] = v;
            }
            for (int i = lane; i < 256; i += 32) P[i] = Pn[i];
            __syncthreads();
        }
    }
}

extern "C" void kernel_launch(void* const* d_in, const int* in_sizes, int n_in,
                              void* d_out, int out_size, void* d_ws, size_t ws_size,
                              hipStream_t stream) {
    (void)in_sizes; (void)n_in; (void)out_size; (void)d_ws; (void)ws_size;
    const float* obs  = (const float*)d_in[0];
    const float* ctrl = (const float*)d_in[1];
    const float* Amat = (const float*)d_in[2];
    const float* Bmat = (const float*)d_in[3];
    const float* Cmat = (const float*)d_in[4];
    const float* Qmat = (const float*)d_in[5];
    const float* Rmat = (const float*)d_in[6];
    const float* boff = (const float*)d_in[7];
    const float* doff = (const float*)d_in[8];
    const float* mu0  = (const float*)d_in[9];
    const float* P0   = (const float*)d_in[10];
    float* out = (float*)d_out;

    KalmanFilter_gfx1250_kernel<<<dim3(NB), dim3(32), 0, stream>>>(
        obs, ctrl, Amat, Bmat, Cmat, Qmat, Rmat, boff, doff, mu0, P0, out);
}